// EdgeAttention_20332375179521
// MI455X (gfx1250) — compile-verified
//
#include <hip/hip_runtime.h>

typedef __attribute__((ext_vector_type(2))) float v2f;
typedef __attribute__((ext_vector_type(8))) float v8f;

#define MCH   256
#define NPX   1024
#define LEAKY 0.01f

// One workgroup per row i. 256 threads = 8 waves.
// Phase 1: w[i,j] = sum_m E[m,i,j]*cw[m] + b ; LeakyReLU ; softmax_j -> LDS
// Phase 2: node[m,i] = sum_j coef[i,j]*E[m,i,j] via broadcast-GEMV on
//          V_WMMA_F32_16X16X4_F32 (every C column holds the same result).
__global__ __launch_bounds__(256) void edge_attn_fused(
    const float* __restrict__ E,   // [256,1024,1024]
    const float* __restrict__ cw,  // [256]
    const float* __restrict__ cb,  // [1]
    float* __restrict__ out)       // [256,1024]
{
  __shared__ float scw[MCH];
  __shared__ float srow[NPX];
  __shared__ float sredm[8];
  __shared__ float sreds[8];

  const int i    = blockIdx.x;
  const int t    = threadIdx.x;
  const int lane = t & 31;
  const int wv   = t >> 5;

  scw[t] = cw[t];
  __syncthreads();

  const size_t rowoff = (size_t)i * NPX;

  // ---------------- Phase 1: channel reduction (coalesced float4 streams) ---
  // Default (RT) temporal hint: this block's 1MB slice should survive in L2
  // until phase 2 re-reads it.
  float4 acc = make_float4(0.f, 0.f, 0.f, 0.f);
  for (int m = 0; m < MCH; ++m) {
    const float4 v =
        *reinterpret_cast<const float4*>(E + (size_t)m * (NPX * NPX) + rowoff + 4 * t);
    const float wm = scw[m];
    acc.x += wm * v.x; acc.y += wm * v.y;
    acc.z += wm * v.z; acc.w += wm * v.w;
  }
  const float bias = cb[0];
  float w0 = acc.x + bias, w1 = acc.y + bias, w2 = acc.z + bias, w3 = acc.w + bias;
  w0 = (w0 >= 0.f) ? w0 : LEAKY * w0;
  w1 = (w1 >= 0.f) ? w1 : LEAKY * w1;
  w2 = (w2 >= 0.f) ? w2 : LEAKY * w2;
  w3 = (w3 >= 0.f) ? w3 : LEAKY * w3;

  // ---------------- softmax over the 1024 values of this row ----------------
  float lmax = fmaxf(fmaxf(w0, w1), fmaxf(w2, w3));
  #pragma unroll
  for (int off = 16; off > 0; off >>= 1)
    lmax = fmaxf(lmax, __shfl_xor(lmax, off, 32));
  if (lane == 0) sredm[wv] = lmax;
  __syncthreads();
  float rmax = sredm[0];
  #pragma unroll
  for (int k = 1; k < 8; ++k) rmax = fmaxf(rmax, sredm[k]);

  const float e0 = __expf(w0 - rmax);
  const float e1 = __expf(w1 - rmax);
  const float e2 = __expf(w2 - rmax);
  const float e3 = __expf(w3 - rmax);
  float lsum = (e0 + e1) + (e2 + e3);
  #pragma unroll
  for (int off = 16; off > 0; off >>= 1)
    lsum += __shfl_xor(lsum, off, 32);
  if (lane == 0) sreds[wv] = lsum;
  __syncthreads();
  float rsum = sreds[0];
  #pragma unroll
  for (int k = 1; k < 8; ++k) rsum += sreds[k];
  const float inv = 1.0f / rsum;

  srow[4 * t + 0] = e0 * inv;
  srow[4 * t + 1] = e1 * inv;
  srow[4 * t + 2] = e2 * inv;
  srow[4 * t + 3] = e3 * inv;
  __syncthreads();

  // ---------------- Phase 2: WMMA broadcast-GEMV over j ---------------------
  // Wave wv owns m-tiles m0 = 16*wv and m0+128. A-fragment (16x4 f32):
  //   lanes 0-15 : rows m, K=0 (a.x) / K=1 (a.y)
  //   lanes16-31 : rows m, K=2 (a.x) / K=3 (a.y)
  // B-fragment (4x16 f32) broadcast: b = coef[i, j+2*half .. j+2*half+1].
  const int half = lane >> 4;       // 0: K=0,1  1: K=2,3
  const int mrow = lane & 15;
  const size_t mstride = (size_t)NPX * NPX;

  const float* e0p = E + (size_t)(wv * 16 + mrow) * mstride + rowoff + 2 * half;
  const float* e1p = e0p + (size_t)128 * mstride;

  v8f c0 = {0.f, 0.f, 0.f, 0.f, 0.f, 0.f, 0.f, 0.f};
  v8f c1 = {0.f, 0.f, 0.f, 0.f, 0.f, 0.f, 0.f, 0.f};

  for (int j = 0; j < NPX; j += 4) {
    if ((j & 31) == 0) {
      // High-locality hint -> WGP-scope prefetch: pull the next cachelines of
      // both row streams into all cache levels (L2 hit expected from phase 1).
      __builtin_prefetch(e0p + j + 128, 0, 3); // -> global_prefetch_b8 (WGP)
      __builtin_prefetch(e1p + j + 128, 0, 3);
    }
    // Last use of this E slice: non-temporal so dead lines don't occupy L2.
    const v2f a0 = __builtin_nontemporal_load(reinterpret_cast<const v2f*>(e0p + j));
    const v2f a1 = __builtin_nontemporal_load(reinterpret_cast<const v2f*>(e1p + j));
    const v2f b  = *reinterpret_cast<const v2f*>(&srow[j + 2 * half]);
    c0 = __builtin_amdgcn_wmma_f32_16x16x4_f32(false, a0, false, b, (short)0, c0,
                                               false, false);
    c1 = __builtin_amdgcn_wmma_f32_16x16x4_f32(false, a1, false, b, (short)0, c1,
                                               false, false);
  }

  // C layout (16x16 f32): VGPR v, lane 0 -> (M=v, N=0); lane 16 -> (M=8+v, N=0).
  // All columns identical, so column 0 is the GEMV result.
  if (lane == 0 || lane == 16) {
    const int mbase = wv * 16 + 8 * half;
    #pragma unroll
    for (int v = 0; v < 8; ++v) {
      out[(size_t)(mbase + v) * NPX + i]       = c0[v];
      out[(size_t)(mbase + 128 + v) * NPX + i] = c1[v];
    }
  }
}

extern "C" void kernel_launch(void* const* d_in, const int* in_sizes, int n_in,
                              void* d_out, int out_size, void* d_ws, size_t ws_size,
                              hipStream_t stream) {
  (void)in_sizes; (void)n_in; (void)out_size; (void)d_ws; (void)ws_size;
  const float* E  = (const float*)d_in[0];  // edge_feats [1,256,1024,1024]
  const float* cw = (const float*)d_in[1];  // conv_w [256]
  const float* cb = (const float*)d_in[2];  // conv_b [1]
  float* out = (float*)d_out;               // node_feats [1,256,1024]
  edge_attn_fused<<<NPX, 256, 0, stream>>>(E, cw, cb, out);
}